// BayesianMF_79422535238262
// MI455X (gfx1250) — compile-verified
//
#include <hip/hip_runtime.h>
#include <hip/hip_bf16.h>

#define DD 16
#define ALPHA 2.0f
#define WPB 8   // waves (users) per block

typedef __attribute__((ext_vector_type(2))) float v2f;
typedef __attribute__((ext_vector_type(8))) float v8f;

// ---------------- pass 1: zero / count / scan / scatter ----------------

__global__ void bmf_zero_kernel(int* __restrict__ p, int n) {
    int i = blockIdx.x * blockDim.x + threadIdx.x;
    if (i < n) p[i] = 0;
}

__global__ void bmf_count_kernel(const int* __restrict__ user_ids,
                                 int* __restrict__ counts, int nnz) {
    int i = blockIdx.x * blockDim.x + threadIdx.x;
    if (i < nnz) atomicAdd(&counts[user_ids[i]], 1);
}

// single block of 1024 threads, chunked sequential exclusive scan
__global__ void bmf_scan_kernel(const int* __restrict__ counts,
                                int* __restrict__ offsets, int n) {
    __shared__ int tsum[1024];
    const int t = threadIdx.x;
    const int chunk = (n + 1023) / 1024;
    const int b = t * chunk;
    const int e = (b + chunk < n) ? (b + chunk) : n;
    int s = 0;
    for (int i = b; i < e; ++i) s += counts[i];
    tsum[t] = s;
    __syncthreads();
    if (t == 0) {
        int run = 0;
        for (int i = 0; i < 1024; ++i) { int v = tsum[i]; tsum[i] = run; run += v; }
        offsets[n] = run;
    }
    __syncthreads();
    int run = tsum[t];
    for (int i = b; i < e; ++i) { offsets[i] = run; run += counts[i]; }
}

__global__ void bmf_copy_kernel(const int* __restrict__ src,
                                int* __restrict__ dst, int n) {
    int i = blockIdx.x * blockDim.x + threadIdx.x;
    if (i < n) dst[i] = src[i];
}

__global__ void bmf_scatter_kernel(const int* __restrict__ user_ids,
                                   int* __restrict__ cursors,
                                   int* __restrict__ sorted, int nnz) {
    int i = blockIdx.x * blockDim.x + threadIdx.x;
    if (i < nnz) {
        int pos = atomicAdd(&cursors[user_ids[i]], 1);
        sorted[pos] = i;
    }
}

// ---------------- pass 2: one wave32 per user ----------------
// WMMA f32 16x16x4 accumulates  sum_j Vj Vj^T  in exact f32.
// A(16x4) layout: lane (k>=2?16:0)+m, vgpr k%2 holds A[m][k] = V[item_k][m]
// B(4x16) layout: same register image with m<->n, so A and B share registers.
// C/D layout: vgpr v, lane L -> (m,n) = (v + 8*(L>>4), L&15)
//
// Batch 32 observations per outer iteration: one coalesced index load +
// 32-wide gathers, distribute per-chunk items/ratings via __shfl, then up to
// 8 back-to-back WMMAs. Chunk-skip branch is wave-uniform so EXEC is all-1s
// at every WMMA.

__global__ __launch_bounds__(256)
void bmf_solve_kernel(const float* __restrict__ V,
                      const float* __restrict__ ratings,
                      const float* __restrict__ mu_K,
                      const float* __restrict__ lambda_K,
                      const float* __restrict__ z,
                      const int* __restrict__ item_ids,
                      const int* __restrict__ offsets,
                      const int* __restrict__ sorted,
                      float* __restrict__ out,
                      int num_users) {
    __shared__ float AugS[WPB][DD][2 * DD + 2];   // 16 x 34 augmented [prec | I]
    __shared__ float RhsS[WPB][DD];

    const int lane = threadIdx.x & 31;
    const int wave = threadIdx.x >> 5;
    const int u = blockIdx.x * WPB + wave;
    if (u >= num_users) return;

    const int beg = offsets[u];
    const int end = offsets[u + 1];

    const int sub = lane >> 4;      // half-wave: items {0,1} or {2,3} of chunk
    const int d   = lane & 15;      // embedding element

    v8f acc = {};                   // 16x16 f32 accumulator (sum Vj Vj^T)
    float ssum = 0.0f;              // half-wave partial of sum r_j * Vj[d]

    for (int base = beg; base < end; base += 32) {
        const int nvalid = end - base;              // wave-uniform
        if (base + 32 < end) __builtin_prefetch(&sorted[base + 32], 0, 1);

        // stage 32 observations wave-wide (one coalesced load + 2 gathers)
        int   item = 0;
        float r    = 0.0f;
        if (lane < nvalid) {
            const int obs = sorted[base + lane];
            item = item_ids[obs];
            r    = ratings[obs];
        }

#pragma unroll
        for (int c = 0; c < 8; ++c) {
            if (4 * c >= nvalid) break;             // uniform: EXEC stays all-1s
            const int p0 = 4 * c + 2 * sub;
            const int p1 = p0 + 1;
            const int   it0 = __shfl(item, p0, 32);
            const int   it1 = __shfl(item, p1, 32);
            float r0 = __shfl(r, p0, 32);
            float r1 = __shfl(r, p1, 32);
            v2f a;
            a.x = (p0 < nvalid) ? V[it0 * DD + d] : 0.0f;
            a.y = (p1 < nvalid) ? V[it1 * DD + d] : 0.0f;
            if (p0 >= nvalid) r0 = 0.0f;
            if (p1 >= nvalid) r1 = 0.0f;
            ssum += r0 * a.x + r1 * a.y;
            // D = A * B + C ; A and B share the same register image
            acc = __builtin_amdgcn_wmma_f32_16x16x4_f32(
                false, a, false, a, (short)0, acc, false, false);
        }
    }
    ssum += __shfl_xor(ssum, 16, 32);   // combine half-waves -> full scoresum[d]

    float (*A)[2 * DD + 2] = AugS[wave];

    // prec = lambda_K + alpha * acc into left half; identity into right half
#pragma unroll
    for (int v = 0; v < 8; ++v) {
        const int m = v + 8 * sub;
        A[m][d]      = lambda_K[m * DD + d] + ALPHA * acc[v];
        A[m][DD + d] = (m == d) ? 1.0f : 0.0f;
    }

    // rhs = alpha * scoresum + lambda_K @ mu_K
    if (lane < DD) {
        float lm = 0.0f;
#pragma unroll
        for (int n = 0; n < DD; ++n) lm += lambda_K[lane * DD + n] * mu_K[n];
        RhsS[wave][lane] = ALPHA * ssum + lm;
    }

    // Gauss-Jordan: lane owns augmented column j; per-wave private tile,
    // wave32 lockstep + in-order LDS => no barriers needed.
    const int j = lane;
    for (int k = 0; k < DD; ++k) {
        const float invp = 1.0f / A[k][k];
        A[k][j] *= invp;
#pragma unroll
        for (int i = 0; i < DD; ++i) {
            if (i == k) continue;
            const float f = A[i][k];           // read before lane k overwrites
            A[i][j] -= f * A[k][j];
        }
    }

    // cov = 0.5 * (inv + inv^T) into left half (lanes 0..15, lane = column)
    if (lane < DD) {
        float tmp[DD];
#pragma unroll
        for (int m = 0; m < DD; ++m)
            tmp[m] = 0.5f * (A[m][DD + lane] + A[lane][DD + m]);
#pragma unroll
        for (int m = 0; m < DD; ++m)
            A[m][lane] = tmp[m];
    }

    // mu = cov @ rhs (before Cholesky destroys cov's lower triangle)
    float mu = 0.0f;
    if (lane < DD) {
#pragma unroll
        for (int n = 0; n < DD; ++n) mu += A[lane][n] * RhsS[wave][n];
    }

    // in-place Cholesky of cov (lower), lane = row
    const int r = lane;
    for (int k = 0; k < DD; ++k) {
        if (lane == 0) A[k][k] = __builtin_sqrtf(A[k][k]);
        const float dk = A[k][k];
        if (r > k && r < DD) A[r][k] /= dk;
        if (r > k && r < DD) {
            const float lrk = A[r][k];
            for (int jj = k + 1; jj <= r; ++jj)
                A[r][jj] -= lrk * A[jj][k];
        }
    }

    // sample: out = mu + L @ z_u   (lower triangle + diag only)
    if (lane < DD) {
        float uacc = mu;
        const float* zu = z + (size_t)u * DD;
#pragma unroll
        for (int e = 0; e < DD; ++e)
            if (e <= lane) uacc += A[lane][e] * zu[e];
        out[(size_t)u * DD + lane] = uacc;
    }
}

// ---------------- launcher ----------------

extern "C" void kernel_launch(void* const* d_in, const int* in_sizes, int n_in,
                              void* d_out, int out_size, void* d_ws, size_t ws_size,
                              hipStream_t stream) {
    const float* V        = (const float*)d_in[0];
    const float* ratings  = (const float*)d_in[1];
    const float* mu_K     = (const float*)d_in[2];
    const float* lambda_K = (const float*)d_in[3];
    const float* z        = (const float*)d_in[4];
    const int*   user_ids = (const int*)d_in[5];
    const int*   item_ids = (const int*)d_in[6];

    const int nnz = in_sizes[1];          // ratings
    const int U   = in_sizes[4] / DD;     // z is [U, 16]
    float* out = (float*)d_out;

    int* counts  = (int*)d_ws;            // U
    int* offsets = counts + U;            // U + 1
    int* cursors = offsets + U + 1;       // U
    int* sorted  = cursors + U;           // nnz

    const int T = 256;
    bmf_zero_kernel<<<(U + T - 1) / T, T, 0, stream>>>(counts, U);
    bmf_count_kernel<<<(nnz + T - 1) / T, T, 0, stream>>>(user_ids, counts, nnz);
    bmf_scan_kernel<<<1, 1024, 0, stream>>>(counts, offsets, U);
    bmf_copy_kernel<<<(U + T - 1) / T, T, 0, stream>>>(offsets, cursors, U);
    bmf_scatter_kernel<<<(nnz + T - 1) / T, T, 0, stream>>>(user_ids, cursors, sorted, nnz);
    bmf_solve_kernel<<<(U + WPB - 1) / WPB, 32 * WPB, 0, stream>>>(
        V, ratings, mu_K, lambda_K, z, item_ids, offsets, sorted, out, U);
}